// SparseConvTranspose_82566451298749
// MI455X (gfx1250) — compile-verified
//
#include <hip/hip_runtime.h>

typedef __attribute__((ext_vector_type(2))) float v2f;
typedef __attribute__((ext_vector_type(8))) float v8f;

#define C_IN          32
#define FILTERS       32
#define NKIDX         27
#define TILE_M        16
#define WAVES_PER_BLK 8
#define PAIRS_PER_BLK (TILE_M * WAVES_PER_BLK)

// Packed B layout: per kidx, [nhalf(2)][kstep(8)][lane(32)] float2,
// where lane = (n & 15) + 16*hi, krow = 4*kstep + 2*hi,
// value = { w[krow][16*nhalf + n], w[krow+1][16*nhalf + n] }.
#define PACK_PER_KIDX (2 * 8 * 32)                  // 512 float2 per kidx
#define PACK_TOTAL    (NKIDX * PACK_PER_KIDX)       // 13824 float2 = 110592 B

// out[i] = bias[i % 32]  (re-initializes the poisoned output buffer each call)
__global__ void sct_bias_init(float* __restrict__ out,
                              const float* __restrict__ bias, int total) {
    int i = blockIdx.x * blockDim.x + threadIdx.x;
    if (i < total) out[i] = bias[i & (FILTERS - 1)];
}

// Swizzle the 27x32x32 weight tensor into the exact WMMA-B VGPR layout so the
// main kernel fetches each B tile as one coalesced b64 load per lane.
__global__ void sct_weight_pack(const float* __restrict__ wflat,  // [27,32,32]
                                v2f* __restrict__ wpacked) {      // [PACK_TOTAL]
    int i = blockIdx.x * blockDim.x + threadIdx.x;
    if (i >= PACK_TOTAL) return;
    int lane = i & 31;
    int s    = (i >> 5) & 7;     // kstep
    int t    = (i >> 8) & 1;     // nhalf
    int kk   = i >> 9;           // kidx
    int row  = lane & 15;
    int hi   = lane >> 4;
    int krow = 4 * s + 2 * hi;
    int col  = 16 * t + row;
    v2f p;
    p.x = wflat[kk * (C_IN * FILTERS) + (krow + 0) * FILTERS + col];
    p.y = wflat[kk * (C_IN * FILTERS) + (krow + 1) * FILTERS + col];
    wpacked[i] = p;
}

// One wave32 handles 16 pairs: C(16x32,f32) += A(16x32 gathered feats) * W[kidx](32x32),
// via v_wmma_f32_16x16x4_f32 (2 N-tiles x 8 K-steps), then scatter-adds rows to out.
template <bool PACKED>
__global__ __launch_bounds__(256)
void sct_pair_wmma(const float* __restrict__ feats,     // [N_in, 32]
                   const float* __restrict__ wflat,     // [27, 32, 32]
                   const v2f*   __restrict__ wpacked,   // [PACK_TOTAL] or unused
                   const float* __restrict__ inp_pos,   // [N_in, 3]
                   const float* __restrict__ out_pos,   // [N_out, 3]
                   const int*   __restrict__ pair_out,  // [E]
                   const int*   __restrict__ pair_inp,  // [E]
                   const float* __restrict__ voxel_p,   // [1]
                   float*       __restrict__ out,       // [N_out, 32]
                   int E) {
    const int lane = threadIdx.x & 31;
    const int wave = threadIdx.x >> 5;
    const int row  = lane & 15;   // M row (and N column) this lane serves
    const int hi   = lane >> 4;   // 0 -> K {0,1}; 1 -> K {2,3} within a K-step

    const int e0 = (blockIdx.x * WAVES_PER_BLK + wave) * TILE_M;
    const int e  = e0 + row;
    const bool valid = (e < E);

    const float inv_ext = 1.0f / (3.0f * voxel_p[0]);

    int pi = 0, po = -1, myk = -1;
    if (valid) {
        pi = pair_inp[e];
        po = pair_out[e];
        float rx = (out_pos[po * 3 + 0] - inp_pos[pi * 3 + 0]) * inv_ext;
        float ry = (out_pos[po * 3 + 1] - inp_pos[pi * 3 + 1]) * inv_ext;
        float rz = (out_pos[po * 3 + 2] - inp_pos[pi * 3 + 2]) * inv_ext;
        int cx = (int)floorf((rx + 0.5f) * 3.0f); cx = min(max(cx, 0), 2);
        int cy = (int)floorf((ry + 0.5f) * 3.0f); cy = min(max(cy, 0), 2);
        int cz = (int)floorf((rz + 0.5f) * 3.0f); cz = min(max(cz, 0), 2);
        myk = (cx * 3 + cy) * 3 + cz;
    }

    v8f c0 = {};  // filters 0..15
    v8f c1 = {};  // filters 16..31

    // A operand, this lane's slice: row `pi`, float2 at K = 4*s + 2*hi.
    const v2f* frv = (const v2f*)(feats + (size_t)pi * C_IN);

    // Iterate distinct kidx values in this tile (usually exactly one, since pairs
    // are emitted per kernel-offset chunk). Wave-uniform loop -> EXEC stays all-ones
    // around WMMA as the ISA requires.
    unsigned long long todo = __ballot(valid) & 0xFFFFull;
    while (todo) {
        int leader = __builtin_ffsll((long long)todo) - 1;
        int kk = __shfl(myk, leader, 32);
        unsigned long long same = __ballot(myk == kk);
        todo &= ~same;
        float rs = ((same >> lane) & 1ull) ? 1.0f : 0.0f;  // zero-mask non-member rows

        const v2f*   wp = wpacked + (size_t)kk * PACK_PER_KIDX;
        const float* wb = wflat + kk * (C_IN * FILTERS);

#pragma unroll
        for (int s = 0; s < 8; ++s) {
            v2f a = frv[2 * s + hi];
            a.x *= rs;
            a.y *= rs;
            v2f b0, b1;
            if (PACKED) {
                b0 = wp[s * 32 + lane];          // coalesced global_load_b64
                b1 = wp[256 + s * 32 + lane];
            } else {
                const int krow = 4 * s + 2 * hi;
                b0.x = wb[(krow + 0) * FILTERS + row];
                b0.y = wb[(krow + 1) * FILTERS + row];
                b1.x = wb[(krow + 0) * FILTERS + 16 + row];
                b1.y = wb[(krow + 1) * FILTERS + 16 + row];
            }
            c0 = __builtin_amdgcn_wmma_f32_16x16x4_f32(
                false, a, false, b0, (short)0, c0, false, false);
            c1 = __builtin_amdgcn_wmma_f32_16x16x4_f32(
                false, a, false, b1, (short)0, c1, false, false);
        }
    }

    // Scatter-add: lane holds C[M = v + 8*hi][N = row (+16)] in c0[v]/c1[v].
#pragma unroll
    for (int v = 0; v < 8; ++v) {
        int M = v + 8 * hi;
        int poM = __shfl(po, M, 32);   // pair_out of row M (ds_bpermute)
        if (poM >= 0) {
            atomicAdd(&out[(size_t)poM * FILTERS + row],      c0[v]);
            atomicAdd(&out[(size_t)poM * FILTERS + 16 + row], c1[v]);
        }
    }
}

extern "C" void kernel_launch(void* const* d_in, const int* in_sizes, int n_in,
                              void* d_out, int out_size, void* d_ws, size_t ws_size,
                              hipStream_t stream) {
    const float* feats = (const float*)d_in[0];
    const float* kern  = (const float*)d_in[1];  // [3,3,3,32,32] == [27,32,32] flat
    const float* bias  = (const float*)d_in[2];
    const float* ipos  = (const float*)d_in[3];
    const float* opos  = (const float*)d_in[4];
    const int*   pout  = (const int*)d_in[5];
    const int*   pinp  = (const int*)d_in[6];
    const float* vox   = (const float*)d_in[7];
    float* out = (float*)d_out;

    const int E = in_sizes[5];

    int init_blocks = (out_size + 255) / 256;
    sct_bias_init<<<init_blocks, 256, 0, stream>>>(out, bias, out_size);

    const size_t pack_bytes = (size_t)PACK_TOTAL * sizeof(v2f);
    const bool use_pack = (ws_size >= pack_bytes);

    int blocks = (E + PAIRS_PER_BLK - 1) / PAIRS_PER_BLK;
    if (use_pack) {
        v2f* wpacked = (v2f*)d_ws;
        sct_weight_pack<<<(PACK_TOTAL + 255) / 256, 256, 0, stream>>>(kern, wpacked);
        if (blocks > 0) {
            sct_pair_wmma<true><<<blocks, 256, 0, stream>>>(
                feats, kern, wpacked, ipos, opos, pout, pinp, vox, out, E);
        }
    } else if (blocks > 0) {
        sct_pair_wmma<false><<<blocks, 256, 0, stream>>>(
            feats, kern, (const v2f*)nullptr, ipos, opos, pout, pinp, vox, out, E);
    }
}